// SparseAttention_11725260718205
// MI455X (gfx1250) — compile-verified
//
#include <hip/hip_runtime.h>

typedef float v2f __attribute__((ext_vector_type(2)));
typedef float v8f __attribute__((ext_vector_type(8)));

#define NBATCH 128
#define NROW   1024
#define DIN    512
#define TOPK_N 12
#define NWAVE  8
#define SPLIT  2      // workgroups per batch (phase-2 strip split)

#define WMMA_F32(A, B, C) \
  __builtin_amdgcn_wmma_f32_16x16x4_f32(false, (A), false, (B), (short)0, (C), false, false)

__device__ __forceinline__ float half16_max(float v) {
  v = fmaxf(v, __shfl_xor(v, 1, 32));
  v = fmaxf(v, __shfl_xor(v, 2, 32));
  v = fmaxf(v, __shfl_xor(v, 4, 32));
  v = fmaxf(v, __shfl_xor(v, 8, 32));
  return v;
}
__device__ __forceinline__ float half16_sum(float v) {
  v += __shfl_xor(v, 1, 32);
  v += __shfl_xor(v, 2, 32);
  v += __shfl_xor(v, 4, 32);
  v += __shfl_xor(v, 8, 32);
  return v;
}

__global__ __launch_bounds__(256, 1)
void zero_ws(float* __restrict__ Aws) {
  Aws[blockIdx.x * 256 + threadIdx.x] = 0.0f;
}

__global__ __launch_bounds__(256, 1)
void sparse_attn_scores(const float* __restrict__ x,
                        const float* __restrict__ wk,
                        const float* __restrict__ wq,
                        float* __restrict__ Aws) {
  // Packed weight pairs padded to all 16 WMMA columns (cols 8..15 = 0):
  // Wp[j][c] = { W[2j][c], W[2j+1][c] },  W = [wk | wq | 0] (512x16)
  __shared__ v2f   Wp[256 * 16];                  // 32 KB
  __shared__ __align__(16) float kqsh[NROW * 8];  // 32 KB: per-row [k0..k3, q0..q3]
  __shared__ float Ash[NROW];                     // 4 KB partial column scores

  const int b     = blockIdx.x >> 1;   // batch
  const int split = blockIdx.x & 1;    // which half of the strips
  const int tid   = threadIdx.x;
  const int lane  = tid & 31;
  const int wave  = tid >> 5;
  const int g     = lane >> 4;         // half-wave: selects K pair / row half
  const int n16   = lane & 15;         // column / row-in-tile index

  // ---- stage packed (padded) weights, zero partial scores ----
  for (int i = tid; i < 4096; i += 256) {
    int j = i >> 4, c = i & 15;
    int k0 = 2 * j;
    v2f w; w.x = 0.0f; w.y = 0.0f;
    if (c < 4)      { w.x = wk[k0 * 4 + c];       w.y = wk[(k0 + 1) * 4 + c]; }
    else if (c < 8) { w.x = wq[k0 * 4 + (c - 4)]; w.y = wq[(k0 + 1) * 4 + (c - 4)]; }
    Wp[i] = w;
  }
  for (int i = tid; i < NROW; i += 256) Ash[i] = 0.0f;
  __syncthreads();

  // ---- phase 1: kq[b] = x[b] * [wk|wq|0], 4 row-tiles per k-step ----
  // (both split-siblings compute the full kq; x[b] is L2-resident)
  for (int grp = 0; grp < 2; ++grp) {
    const int tbase = wave * 8 + grp * 4;
    const float* xp = x + ((size_t)b * NROW + tbase * 16 + n16) * DIN + 2 * g;
    if (grp == 0)
      __builtin_prefetch(xp + 4 * 16 * DIN, 0, 1);  // warm next group
    v8f c0 = {}, c1 = {}, c2 = {}, c3 = {};
    #pragma unroll 2
    for (int kk = 0; kk < DIN / 4; ++kk) {
      v2f bb = Wp[(2 * kk + g) * 16 + n16];         // shared across 4 tiles
      v2f a0 = *(const v2f*)(xp + 0 * 16 * DIN + 4 * kk);
      v2f a1 = *(const v2f*)(xp + 1 * 16 * DIN + 4 * kk);
      v2f a2 = *(const v2f*)(xp + 2 * 16 * DIN + 4 * kk);
      v2f a3 = *(const v2f*)(xp + 3 * 16 * DIN + 4 * kk);
      c0 = WMMA_F32(a0, bb, c0);
      c1 = WMMA_F32(a1, bb, c1);
      c2 = WMMA_F32(a2, bb, c2);
      c3 = WMMA_F32(a3, bb, c3);
    }
    if (n16 < 8) {
      #pragma unroll
      for (int v = 0; v < 8; ++v) {
        int r = tbase * 16 + v + 8 * g;
        kqsh[(r +  0) * 8 + n16] = c0[v];
        kqsh[(r + 16) * 8 + n16] = c1[v];
        kqsh[(r + 32) * 8 + n16] = c2[v];
        kqsh[(r + 48) * 8 + n16] = c3[v];
      }
    }
  }
  __syncthreads();

  // ---- phase 2: column scores, 3 WMMA sweeps, reductions hoisted out ----
  // exp(s*h) == exp2(s*log2e*h): fold log2e into the scale, use native v_exp
  const float scale2 = 0.044194173824159216f * 1.4426950408889634f;
  {
    const int sbase = split * 32 + wave * 4;   // 4 consecutive strips per wave
    v2f a[4];
    #pragma unroll
    for (int st = 0; st < 4; ++st)
      a[st] = *(const v2f*)(&kqsh[((sbase + st) * 16 + n16) * 8 + 2 * g]);

    // sweep 1: per-lane running raw max (1 v_max per row per block, no shuffles)
    float mx[4][8];
    #pragma unroll
    for (int st = 0; st < 4; ++st)
      #pragma unroll
      for (int v = 0; v < 8; ++v) mx[st][v] = -3.0e38f;
    for (int j = 0; j < NROW / 16; ++j) {
      v2f bq = *(const v2f*)(&kqsh[(j * 16 + n16) * 8 + 4 + 2 * g]);
      v8f c[4];
      #pragma unroll
      for (int st = 0; st < 4; ++st) {
        v8f z = {};
        c[st] = WMMA_F32(a[st], bq, z);
        #pragma unroll
        for (int v = 0; v < 8; ++v) mx[st][v] = fmaxf(mx[st][v], c[st][v]);
      }
    }
    #pragma unroll
    for (int st = 0; st < 4; ++st)
      #pragma unroll
      for (int v = 0; v < 8; ++v)
        mx[st][v] = half16_max(mx[st][v]) * scale2;   // final log2-domain row max

    // sweep 2: per-lane partial exp sums (fma + exp + add per row per block)
    float acc[4][8];
    #pragma unroll
    for (int st = 0; st < 4; ++st)
      #pragma unroll
      for (int v = 0; v < 8; ++v) acc[st][v] = 0.0f;
    for (int j = 0; j < NROW / 16; ++j) {
      v2f bq = *(const v2f*)(&kqsh[(j * 16 + n16) * 8 + 4 + 2 * g]);
      v8f c[4];
      #pragma unroll
      for (int st = 0; st < 4; ++st) {
        v8f z = {};
        c[st] = WMMA_F32(a[st], bq, z);
        #pragma unroll
        for (int v = 0; v < 8; ++v)
          acc[st][v] += __builtin_amdgcn_exp2f(fmaf(c[st][v], scale2, -mx[st][v]));
      }
    }
    #pragma unroll
    for (int st = 0; st < 4; ++st)
      #pragma unroll
      for (int v = 0; v < 8; ++v)
        acc[st][v] = 1.0f / half16_sum(acc[st][v]);   // now 1/Z, replicated

    // sweep 3: accumulate normalized probabilities of all 4 strips into A[m]
    for (int j = 0; j < NROW / 16; ++j) {
      v2f bq = *(const v2f*)(&kqsh[(j * 16 + n16) * 8 + 4 + 2 * g]);
      float cs = 0.0f;
      #pragma unroll
      for (int st = 0; st < 4; ++st) {
        v8f z = {};
        v8f c = WMMA_F32(a[st], bq, z);
        #pragma unroll
        for (int v = 0; v < 8; ++v)
          cs += __builtin_amdgcn_exp2f(fmaf(c[v], scale2, -mx[st][v])) * acc[st][v];
      }
      cs += __shfl_xor(cs, 16, 32);                   // merge row halves
      if (lane < 16) atomicAdd(&Ash[j * 16 + n16], cs);
    }
  }
  __syncthreads();

  // ---- merge this WG's partial column scores into the global accumulator ----
  for (int i = tid; i < NROW; i += 256)
    atomicAdd(&Aws[(size_t)b * NROW + i], Ash[i]);
}

__global__ __launch_bounds__(32, 1)
void topk_kernel(const float* __restrict__ Aws, int* __restrict__ out) {
  __shared__ float As[NROW];
  const int b = blockIdx.x;
  const int lane = threadIdx.x;
  for (int i = lane; i < NROW; i += 32) As[i] = Aws[(size_t)b * NROW + i];
  __syncthreads();
  for (int it = 0; it < TOPK_N; ++it) {
    float bv = -1.0e30f;
    int   bi = 0;
    for (int m = lane; m < NROW; m += 32) {
      float v = As[m];
      if (v > bv) { bv = v; bi = m; }       // ascending m keeps lowest tie index
    }
    #pragma unroll
    for (int off = 16; off >= 1; off >>= 1) {
      float ov = __shfl_xor(bv, off, 32);
      int   oi = __shfl_xor(bi, off, 32);
      if (ov > bv || (ov == bv && oi < bi)) { bv = ov; bi = oi; }
    }
    if (lane == 0) {
      out[b * TOPK_N + it] = bi;
      As[bi] = -1.0e30f;
    }
    __builtin_amdgcn_wave_barrier();
  }
}

extern "C" void kernel_launch(void* const* d_in, const int* in_sizes, int n_in,
                              void* d_out, int out_size, void* d_ws, size_t ws_size,
                              hipStream_t stream) {
  const float* x  = (const float*)d_in[0];
  const float* wk = (const float*)d_in[1];
  const float* wq = (const float*)d_in[2];
  float* Aws = (float*)d_ws;               // 128*1024 f32 = 512 KB scratch
  int*   out = (int*)d_out;                // reference output: int32 top-k indices

  zero_ws<<<dim3(NBATCH * NROW / 256), dim3(256), 0, stream>>>(Aws);
  sparse_attn_scores<<<dim3(NBATCH * SPLIT), dim3(256), 0, stream>>>(x, wk, wq, Aws);
  topk_kernel<<<dim3(NBATCH), dim3(32), 0, stream>>>(Aws, out);
}